// MultiHeadPredictor_75376676044989
// MI455X (gfx1250) — compile-verified
//
#include <hip/hip_runtime.h>
#include <math.h>

// ---------------------------------------------------------------------------
// MultiHeadPredictor for MI455X (gfx1250): fp32 WMMA GEMMs + deterministic
// InstanceNorm + LDS-staged brute-force KNN + decomposed EdgeConv max-scatter.
// ---------------------------------------------------------------------------

#define N_TOT   16384
#define N_GR    4
#define NPG     4096          // nodes per graph (N_TOT / N_GR)
#define C_IN    64
#define H_DIM   256
#define H_HALF  128
#define C_OUT   17
#define KNN_K   4
#define RS      8             // row-splits for IN partial sums
#define EPS_IN  1e-5f
#define NEG_SL  0.01f

typedef float v2f __attribute__((ext_vector_type(2)));
typedef float v8f __attribute__((ext_vector_type(8)));

// order-preserving float <-> uint mapping (exact max via integer atomics)
__device__ __forceinline__ unsigned encf(float f) {
  unsigned u = __float_as_uint(f);
  return (u & 0x80000000u) ? ~u : (u | 0x80000000u);
}
__device__ __forceinline__ float decf(unsigned e) {
  unsigned u = (e & 0x80000000u) ? (e & 0x7FFFFFFFu) : ~e;
  return __uint_as_float(u);
}

// ---------------------------------------------------------------------------
// GEMM: Y[N_TOT,Co] = X[N_TOT,Ci] @ (Wa - Wb)[Ci,Co] + bias  (Wb/bias optional)
// Block: 8 waves x 16 rows, TWO 16-column panels (32 cols). Each wave owns one
// 16-row tile and both panels -> one A fragment feeds two WMMAs (halves the
// A-operand L2 traffic, doubles WMMA density per loop iteration).
//   A frag (16x4 f32, 2 VGPR): lanes 0-15 rows M, VGPR0/1 = K{0,1}; lanes
//   16-31 same rows, K{2,3}  -> one aligned b64 load per lane per K-step.
//   B frag (4x16): VGPR0 = row k (lo half) / k+2 (hi half), cols across lanes.
//   C/D  : VGPR v -> row v (lo half) / v+8 (hi half), col = lane&15.
// The two W panels are separate 16-float-stride LDS arrays: 16-stride keeps
// the two half-waves (rows k vs k+2) on disjoint bank sets -> conflict-free.
// ---------------------------------------------------------------------------
#define PANEL (H_DIM * 16)

__global__ __launch_bounds__(256)
void gemm_wmma_f32(const float* __restrict__ X, const float* __restrict__ Wa,
                   const float* __restrict__ Wb, const float* __restrict__ bias,
                   float* __restrict__ Y, int Ci, int Co) {
  __shared__ float wtile[2 * PANEL];           // 2 panels, 16 KB each

  const int col0 = blockIdx.x * 32;
  // cooperative panel load (optionally fused Wa - Wb for the EdgeConv split)
  for (int idx = threadIdx.x; idx < Ci * 32; idx += 256) {
    int k = idx >> 5, cc = idx & 31;           // cc: column within the 32
    int c = col0 + cc;
    float w = Wa[k * Co + c];
    if (Wb) w -= Wb[k * Co + c];
    wtile[(cc >> 4) * PANEL + k * 16 + (cc & 15)] = w;
  }
  __syncthreads();

  const int lane  = threadIdx.x & 31;
  const int wave  = threadIdx.x >> 5;
  const int row0  = (blockIdx.y * 8 + wave) * 16;
  const int mrow  = lane & 15;
  const int khalf = lane >> 4;                 // 0: K{0,1}, 1: K{2,3}
  const int ncol  = lane & 15;

  const float* xrow = X + (size_t)(row0 + mrow) * Ci + 2 * khalf;

  v8f acc0 = {}, acc1 = {};
  for (int k0 = 0; k0 < Ci; k0 += 4) {
    v2f a = *(const v2f*)(xrow + k0);          // global_load_b64
    __builtin_prefetch(xrow + k0 + 32, 0, 0);  // global_prefetch_b8
    int kk = k0 + 2 * khalf;
    v2f b0, b1;
    b0.x = wtile[kk * 16 + ncol];              // ds_load (panel 0)
    b0.y = wtile[(kk + 1) * 16 + ncol];
    b1.x = wtile[PANEL + kk * 16 + ncol];      // ds_load (panel 1)
    b1.y = wtile[PANEL + (kk + 1) * 16 + ncol];
    acc0 = __builtin_amdgcn_wmma_f32_16x16x4_f32(
        false, a, false, b0, (short)0, acc0, false, false);
    acc1 = __builtin_amdgcn_wmma_f32_16x16x4_f32(
        false, a, false, b1, (short)0, acc1, false, false);
  }

  float bb0 = bias ? bias[col0 + ncol]      : 0.0f;
  float bb1 = bias ? bias[col0 + 16 + ncol] : 0.0f;
  union { v8f v; float f[8]; } u0, u1; u0.v = acc0; u1.v = acc1;
#pragma unroll
  for (int v = 0; v < 8; ++v) {
    int r = row0 + v + 8 * khalf;
    Y[(size_t)r * Co + col0 + ncol]      = u0.f[v] + bb0;
    Y[(size_t)r * Co + col0 + 16 + ncol] = u1.f[v] + bb1;
  }
}

// ---------------------------------------------------------------------------
// InstanceNorm: deterministic three-stage reduction. Stage 1 writes RS partial
// (sum, sum2) per (graph, channel) in fixed order; stage 2 finalizes RS
// partials (fixed order) into mean / rsqrt(var+eps); stage 3 applies the
// standardization + LeakyReLU in place at streaming rate.
// ---------------------------------------------------------------------------
__global__ __launch_bounds__(256)
void in_stats(const float* __restrict__ Y, float* __restrict__ pS,
              float* __restrict__ pS2, int C) {
  const int cb = blockIdx.x, g = blockIdx.y, z = blockIdx.z;
  const int cl = threadIdx.x & 63, rl = threadIdx.x >> 6;
  const int c  = cb * 64 + cl;
  const int r0 = z * (NPG / RS);
  float s = 0.f, s2 = 0.f;
  for (int r = r0 + rl; r < r0 + NPG / RS; r += 4) {
    float v = Y[(size_t)(g * NPG + r) * C + c];
    s += v; s2 += v * v;
  }
  __shared__ float la[4][64], lb[4][64];
  la[rl][cl] = s; lb[rl][cl] = s2;
  __syncthreads();
  if (rl == 0) {
    float ts = la[0][cl] + la[1][cl] + la[2][cl] + la[3][cl];
    float t2 = lb[0][cl] + lb[1][cl] + lb[2][cl] + lb[3][cl];
    pS [(size_t)(z * N_GR + g) * C + c] = ts;
    pS2[(size_t)(z * N_GR + g) * C + c] = t2;
  }
}

__global__ __launch_bounds__(256)
void in_finalize(const float* __restrict__ pS, const float* __restrict__ pS2,
                 float* __restrict__ mean, float* __restrict__ inv, int C) {
  const int idx = blockIdx.x * blockDim.x + threadIdx.x;   // over G*C
  if (idx >= N_GR * C) return;
  const int g = idx / C, c = idx % C;
  float s = 0.f, s2 = 0.f;
#pragma unroll
  for (int z = 0; z < RS; ++z) {
    s  += pS [(size_t)(z * N_GR + g) * C + c];
    s2 += pS2[(size_t)(z * N_GR + g) * C + c];
  }
  const float inv_n = 1.0f / (float)NPG;
  float m   = s * inv_n;
  float var = s2 * inv_n - m * m;
  mean[idx] = m;
  inv[idx]  = rsqrtf(var + EPS_IN);
}

__global__ __launch_bounds__(256)
void in_apply(float* __restrict__ Y, const float* __restrict__ mean,
              const float* __restrict__ inv, int C) {
  const size_t idx = (size_t)blockIdx.x * blockDim.x + threadIdx.x;
  if (idx >= (size_t)N_TOT * C) return;
  const int row = (int)(idx / C), c = (int)(idx % C);
  const int g = row / NPG;
  float v = (Y[idx] - mean[g * C + c]) * inv[g * C + c];
  Y[idx] = v >= 0.f ? v : NEG_SL * v;
}

// ---------------------------------------------------------------------------
// Offset head: offset = h2 @ w_o3 + b_o3 ; pos_offset = pos + offset
// ---------------------------------------------------------------------------
__global__ __launch_bounds__(256)
void offset_head(const float* __restrict__ h2, const float* __restrict__ w,
                 const float* __restrict__ b, const float* __restrict__ pos,
                 float* __restrict__ off_out, float* __restrict__ posoff) {
  const int i = blockIdx.x * blockDim.x + threadIdx.x;
  if (i >= N_TOT) return;
  float o0 = b[0], o1 = b[1], o2 = b[2];
  const float* hr = h2 + (size_t)i * H_HALF;
  for (int k = 0; k < H_HALF; ++k) {
    float h = hr[k];
    o0 = fmaf(h, w[k * 3 + 0], o0);
    o1 = fmaf(h, w[k * 3 + 1], o1);
    o2 = fmaf(h, w[k * 3 + 2], o2);
  }
  off_out[i * 3 + 0] = o0; off_out[i * 3 + 1] = o1; off_out[i * 3 + 2] = o2;
  posoff[i * 3 + 0] = pos[i * 3 + 0] + o0;
  posoff[i * 3 + 1] = pos[i * 3 + 1] + o1;
  posoff[i * 3 + 2] = pos[i * 3 + 2] + o2;
}

// ---------------------------------------------------------------------------
// Brute-force KNN per graph; whole graph's positions staged in LDS (48 KB,
// well under the 320 KB WGP budget). The j-scan reads are wave-wide LDS
// broadcasts. Strict '<' insertion keeps the first index on ties (matches
// lax.top_k); the self node (d2 == 0) is naturally included, per reference.
// ---------------------------------------------------------------------------
__global__ __launch_bounds__(256)
void knn_kernel(const float* __restrict__ posoff, int* __restrict__ nbr) {
  __shared__ float sx[NPG], sy[NPG], sz[NPG];
  const int g = blockIdx.y;
  for (int j = threadIdx.x; j < NPG; j += blockDim.x) {
    const float* p = posoff + (size_t)(g * NPG + j) * 3;
    sx[j] = p[0]; sy[j] = p[1]; sz[j] = p[2];
  }
  __syncthreads();
  const int il = blockIdx.x * blockDim.x + threadIdx.x;
  const float px = sx[il], py = sy[il], pz = sz[il];
  float d0 = INFINITY, d1 = INFINITY, d2_ = INFINITY, d3 = INFINITY;
  int   i0 = 0, i1 = 0, i2 = 0, i3 = 0;
  for (int j = 0; j < NPG; ++j) {
    float dx = sx[j] - px, dy = sy[j] - py, dz = sz[j] - pz;
    float d = dx * dx + dy * dy + dz * dz;
    if (d < d3) {
      if (d < d0)      { d3=d2_; i3=i2; d2_=d1; i2=i1; d1=d0; i1=i0; d0=d; i0=j; }
      else if (d < d1) { d3=d2_; i3=i2; d2_=d1; i2=i1; d1=d;  i1=j; }
      else if (d < d2_){ d3=d2_; i3=i2; d2_=d;  i2=j; }
      else             { d3=d;   i3=j; }
    }
  }
  int* o = nbr + (size_t)(g * NPG + il) * KNN_K;
  o[0] = g * NPG + i0; o[1] = g * NPG + i1;
  o[2] = g * NPG + i2; o[3] = g * NPG + i3;
}

// ---------------------------------------------------------------------------
// EdgeConv aggregation.  msg = U[dst] + V[src]; undirected knn edges mean
// node i receives V[j] for j in nbr(i) and V[i] flows to each j in nbr(i).
// Exact deterministic float max via monotone-uint atomicMax.
// ---------------------------------------------------------------------------
__global__ __launch_bounds__(256)
void agg_init(unsigned* __restrict__ agg) {
  size_t idx = (size_t)blockIdx.x * blockDim.x + threadIdx.x;
  if (idx < (size_t)N_TOT * H_DIM) agg[idx] = 0u;   // < enc(-inf)
}

__global__ __launch_bounds__(256)
void scatter_max(const float* __restrict__ V, const int* __restrict__ nbr,
                 unsigned* __restrict__ agg) {
  size_t idx = (size_t)blockIdx.x * blockDim.x + threadIdx.x;
  if (idx >= (size_t)N_TOT * H_DIM) return;
  const int i = (int)(idx >> 8);                    // H_DIM == 256
  const int c = (int)(idx & 255);
  const unsigned evi = encf(V[idx]);
#pragma unroll
  for (int t = 0; t < KNN_K; ++t) {
    int j = nbr[i * KNN_K + t];
    atomicMax(&agg[(size_t)i * H_DIM + c], encf(V[(size_t)j * H_DIM + c]));
    atomicMax(&agg[(size_t)j * H_DIM + c], evi);
  }
}

__global__ __launch_bounds__(256)
void combine_uv(const float* __restrict__ U, const unsigned* __restrict__ agg,
                float* __restrict__ out) {
  size_t idx = (size_t)blockIdx.x * blockDim.x + threadIdx.x;
  if (idx < (size_t)N_TOT * H_DIM) out[idx] = U[idx] + decf(agg[idx]);
}

// ---------------------------------------------------------------------------
// Final head: logits = s @ w_s4 + b_s4 ; softmax over 17 classes.
// ---------------------------------------------------------------------------
__global__ __launch_bounds__(256)
void head_softmax(const float* __restrict__ s, const float* __restrict__ w,
                  const float* __restrict__ b, float* __restrict__ probs) {
  const int i = blockIdx.x * blockDim.x + threadIdx.x;
  if (i >= N_TOT) return;
  float l[C_OUT];
#pragma unroll
  for (int c = 0; c < C_OUT; ++c) l[c] = b[c];
  const float* sr = s + (size_t)i * H_HALF;
  for (int k = 0; k < H_HALF; ++k) {
    float v = sr[k];
#pragma unroll
    for (int c = 0; c < C_OUT; ++c) l[c] = fmaf(v, w[k * C_OUT + c], l[c]);
  }
  float mx = l[0];
#pragma unroll
  for (int c = 1; c < C_OUT; ++c) mx = fmaxf(mx, l[c]);
  float sum = 0.f;
#pragma unroll
  for (int c = 0; c < C_OUT; ++c) { l[c] = __expf(l[c] - mx); sum += l[c]; }
  float inv = 1.0f / sum;
#pragma unroll
  for (int c = 0; c < C_OUT; ++c) probs[(size_t)i * C_OUT + c] = l[c] * inv;
}

// ---------------------------------------------------------------------------
extern "C" void kernel_launch(void* const* d_in, const int* in_sizes, int n_in,
                              void* d_out, int out_size, void* d_ws, size_t ws_size,
                              hipStream_t stream) {
  const float* x    = (const float*)d_in[0];
  const float* pos  = (const float*)d_in[1];
  // d_in[2] edge_index, d_in[3] batch: unused (reference ignores edge_index;
  // batch structure is fixed: 4 equal sorted graphs)
  const float* w_o1 = (const float*)d_in[4];  const float* b_o1 = (const float*)d_in[5];
  const float* w_o2 = (const float*)d_in[6];  const float* b_o2 = (const float*)d_in[7];
  const float* w_o3 = (const float*)d_in[8];  const float* b_o3 = (const float*)d_in[9];
  const float* w_s1 = (const float*)d_in[10]; const float* b_s1 = (const float*)d_in[11];
  const float* w_e  = (const float*)d_in[12]; const float* b_e  = (const float*)d_in[13];
  const float* w_s3 = (const float*)d_in[14]; const float* b_s3 = (const float*)d_in[15];
  const float* w_s4 = (const float*)d_in[16]; const float* b_s4 = (const float*)d_in[17];

  float* probs   = (float*)d_out;                       // [N,17]
  float* off_out = probs + (size_t)N_TOT * C_OUT;       // [N,3]

  // workspace carve-up (~67 MB)
  float*    buf0   = (float*)d_ws;                      // [N,256]
  float*    buf1   = buf0 + (size_t)N_TOT * H_DIM;      // [N,256]
  float*    buf2   = buf1 + (size_t)N_TOT * H_DIM;      // [N,256]
  unsigned* agg    = (unsigned*)(buf2 + (size_t)N_TOT * H_DIM); // [N,256]
  float*    pS     = (float*)(agg + (size_t)N_TOT * H_DIM);     // [RS,G,256]
  float*    pS2    = pS   + RS * N_GR * H_DIM;
  float*    mean   = pS2  + RS * N_GR * H_DIM;          // [G,256]
  float*    inv    = mean + N_GR * H_DIM;               // [G,256]
  float*    posoff = inv  + N_GR * H_DIM;               // [N,3]
  int*      nbr    = (int*)(posoff + (size_t)N_TOT * 3);// [N,4]

  const dim3 blk(256);
  const int rowBlocks = N_TOT / 128;                    // 8 wave-tiles per block

#define RUN_IN(BUF, C)                                                         \
  in_stats<<<dim3((C)/64, N_GR, RS), blk, 0, stream>>>(BUF, pS, pS2, C);       \
  in_finalize<<<(N_GR*(C)+255)/256, blk, 0, stream>>>(pS, pS2, mean, inv, C);  \
  in_apply<<<(N_TOT*(C))/256, blk, 0, stream>>>(BUF, mean, inv, C)

  // ---- offset head -------------------------------------------------------
  gemm_wmma_f32<<<dim3(H_DIM/32, rowBlocks), blk, 0, stream>>>(x, w_o1, nullptr, b_o1, buf0, C_IN, H_DIM);
  RUN_IN(buf0, H_DIM);

  gemm_wmma_f32<<<dim3(H_HALF/32, rowBlocks), blk, 0, stream>>>(buf0, w_o2, nullptr, b_o2, buf1, H_DIM, H_HALF);
  RUN_IN(buf1, H_HALF);

  offset_head<<<N_TOT/256, blk, 0, stream>>>(buf1, w_o3, b_o3, pos, off_out, posoff);

  // ---- knn graph on offset positions ------------------------------------
  knn_kernel<<<dim3(NPG/256, N_GR), blk, 0, stream>>>(posoff, nbr);

  // ---- semantic head -----------------------------------------------------
  gemm_wmma_f32<<<dim3(H_DIM/32, rowBlocks), blk, 0, stream>>>(x, w_s1, nullptr, b_s1, buf2, C_IN, H_DIM);
  RUN_IN(buf2, H_DIM);

  // EdgeConv split: U = s1 @ (W_top - W_bot) + b_e,  V = s1 @ W_bot
  const float* w_e_bot = w_e + (size_t)H_DIM * H_DIM;   // rows 256..511
  gemm_wmma_f32<<<dim3(H_DIM/32, rowBlocks), blk, 0, stream>>>(buf2, w_e, w_e_bot, b_e, buf0, H_DIM, H_DIM);
  gemm_wmma_f32<<<dim3(H_DIM/32, rowBlocks), blk, 0, stream>>>(buf2, w_e_bot, nullptr, nullptr, buf1, H_DIM, H_DIM);

  agg_init<<<(N_TOT*H_DIM)/256, blk, 0, stream>>>(agg);
  scatter_max<<<(N_TOT*H_DIM)/256, blk, 0, stream>>>(buf1, nbr, agg);
  combine_uv<<<(N_TOT*H_DIM)/256, blk, 0, stream>>>(buf0, agg, buf2);

  RUN_IN(buf2, H_DIM);

  gemm_wmma_f32<<<dim3(H_HALF/32, rowBlocks), blk, 0, stream>>>(buf2, w_s3, nullptr, b_s3, buf1, H_DIM, H_HALF);
  RUN_IN(buf1, H_HALF);

  head_softmax<<<N_TOT/256, blk, 0, stream>>>(buf1, w_s4, b_s4, probs);
#undef RUN_IN
}